// SlaterPooling_36429912605031
// MI455X (gfx1250) — compile-verified
//
#include <hip/hip_runtime.h>
#include <stdint.h>

#define NMO   40
#define NUP   8
#define NELEC 16

typedef uint32_t u32x4 __attribute__((ext_vector_type(4)));
typedef uint32_t u32x8 __attribute__((ext_vector_type(8)));

__global__ __launch_bounds__(256) void slater_det_kernel(
    const float* __restrict__ mo, const int* __restrict__ cup,
    const int* __restrict__ cdown, float* __restrict__ out, int nconf)
{
    // LDS tile: mo[b] = 16 electrons x 40 MOs, 2560 bytes.
    __shared__ float smo[NELEC * NMO];

    const int b = blockIdx.x;
    const int c = threadIdx.x;

    // Low 32 bits of a flat LDS pointer == wave-relative LDS byte offset
    // Also escapes smo so the asm
    // "memory" clobber provably covers it.
    const uint32_t lds_off = (uint32_t)(uintptr_t)(&smo[0]);

    // ---- Issue TDM DMA of mo[b,:,:] (contiguous 2560 B) ASAP (wave 0) ----
    if (threadIdx.x < 32) {
        uint64_t ga = (uint64_t)(uintptr_t)(mo + (size_t)b * (NELEC * NMO));
        u32x4 g0; u32x8 g1; u32x4 g2; u32x4 g3;
        // D# group 0: count=1 (valid), lds_addr, global_addr[56:0], type=2
        g0[0] = 1u;
        g0[1] = lds_off;
        g0[2] = (uint32_t)ga;
        g0[3] = (uint32_t)((ga >> 32) & 0x1FFFFFFull) | (2u << 30);
        // D# group 1: data_size=4B; tensor 40x16; tile 40x16; stride 40;
        //             no multicast / barrier / iterate / padding
        g1[0] = (2u << 16);                        // wg_mask=0 | data_size=2 (4B)
        g1[1] = ((uint32_t)NMO   & 0xFFFFu) << 16; // tensor_dim0[15:0]=40
        g1[2] = ((uint32_t)NELEC & 0xFFFFu) << 16; // dim0 hi=0 | tensor_dim1=16
        g1[3] = ((uint32_t)NMO   & 0xFFFFu) << 16; // dim1 hi=0 | tile_dim0=40
        g1[4] = (uint32_t)NELEC;                   // tile_dim1=16 | tile_dim2=0
        g1[5] = (uint32_t)NMO;                     // tensor_dim0_stride=40
        g1[6] = 0u;
        g1[7] = 0u;
        g2[0] = g2[1] = g2[2] = g2[3] = 0u;        // dims 2/3 unused
        g3[0] = g3[1] = g3[2] = g3[3] = 0u;        // dim 4 unused
        asm volatile("tensor_load_to_lds %0, %1, %2, %3"
                     :: "s"(g0), "s"(g1), "s"(g2), "s"(g3) : "memory");
    }

    // ---- Overlap: every thread fetches its config rows while the DMA flies ----
    int colsU[NUP], colsD[NUP];
    #pragma unroll
    for (int j = 0; j < NUP; ++j) {
        colsU[j] = cup[c * NUP + j];
        colsD[j] = cdown[c * NUP + j];
    }

    if (threadIdx.x < 32) {
        __builtin_amdgcn_s_wait_tensorcnt(0);
    }
    __syncthreads();

    // Fully unrolled, register-resident 8x8 LU with partial pivoting.
    // Pivot swap compares the runtime pivot index against unrolled row ids
    // (selects only) so every matrix index stays compile-time -> pure VGPRs.
    auto det8 = [&](int base, const int* cols) -> float {
        float a[8][8];
        #pragma unroll
        for (int i = 0; i < 8; ++i) {
            #pragma unroll
            for (int j = 0; j < 8; ++j)
                a[i][j] = smo[base + i * NMO + cols[j]];  // conflict-free ds_load
        }
        float det = 1.0f;
        #pragma unroll
        for (int k = 0; k < 8; ++k) {
            // pivot search in column k
            int p = k;
            float amax = __builtin_fabsf(a[k][k]);
            #pragma unroll
            for (int r = 0; r < 8; ++r) {
                if (r > k) {
                    float v = __builtin_fabsf(a[r][k]);
                    if (v > amax) { amax = v; p = r; }
                }
            }
            // conditional row swap (static indices, runtime predicate)
            #pragma unroll
            for (int r = 0; r < 8; ++r) {
                if (r > k) {
                    bool sw = (r == p);
                    #pragma unroll
                    for (int j = 0; j < 8; ++j) {
                        if (j >= k) {
                            float tk = a[k][j], tr = a[r][j];
                            a[k][j] = sw ? tr : tk;
                            a[r][j] = sw ? tk : tr;
                        }
                    }
                    det = sw ? -det : det;
                }
            }
            float piv = a[k][k];
            det *= piv;
            float inv = 1.0f / piv;
            #pragma unroll
            for (int r = 0; r < 8; ++r) {
                if (r > k) {
                    float f = a[r][k] * inv;
                    #pragma unroll
                    for (int j = 0; j < 8; ++j) {
                        if (j > k)
                            a[r][j] = __builtin_fmaf(-f, a[k][j], a[r][j]);
                    }
                }
            }
        }
        return det;
    };

    float du = det8(0, colsU);            // electrons 0..7  (spin up)
    float dd = det8(NUP * NMO, colsD);    // electrons 8..15 (spin down)

    out[(size_t)b * nconf + c] = du * dd; // coalesced f32 store
}

extern "C" void kernel_launch(void* const* d_in, const int* in_sizes, int n_in,
                              void* d_out, int out_size, void* d_ws, size_t ws_size,
                              hipStream_t stream) {
    const float* mo    = (const float*)d_in[0];
    const int*   cup   = (const int*)d_in[1];
    const int*   cdown = (const int*)d_in[2];
    float*       out   = (float*)d_out;

    const int nbatch = in_sizes[0] / (NELEC * NMO);  // 2048
    const int nconf  = in_sizes[1] / NUP;            // 256

    slater_det_kernel<<<nbatch, nconf, 0, stream>>>(mo, cup, cdown, out, nconf);
}